// DQSEncoder_61770219651233
// MI455X (gfx1250) — compile-verified
//
#include <hip/hip_runtime.h>

typedef __attribute__((ext_vector_type(2))) float v2f;
typedef __attribute__((ext_vector_type(8))) float v8f;

#define N_NODES 2000000
#define KDIM    20     // 2 * N_MODES
#define NMODES  10

// qm4 is the 20x4 matrix: rows 0..9 = x.reshape(10,4), rows 10..19 = (0,0,0,1)
__device__ __forceinline__ float qm4_elem(const float* __restrict__ x, int k, int c) {
    if (k < NMODES) return x[4 * k + c];
    return (c == 3) ? 1.0f : 0.0f;
}

__global__ __launch_bounds__(256)
void dqs_wmma_kernel(const float* __restrict__ x,
                     const float* __restrict__ weights,
                     const float* __restrict__ VR,
                     float* __restrict__ out)
{
    // per-wave staging: 32 nodes x 4 quaternion components
    __shared__ float qlds[8][32][4];

    const int  lane = threadIdx.x & 31;
    const int  wave = threadIdx.x >> 5;
    const int  col  = lane & 15;       // N index within 16x16 tile
    const bool hi   = lane >= 16;

    const int  nodeBase = blockIdx.x * 256 + wave * 32;  // wave-uniform
    const bool active   = nodeBase < N_NODES;            // wave-uniform -> EXEC all-ones

    v8f c0 = {};
    v8f c1 = {};

    if (active) {
        // ---- B fragments: qm4 sliced into 5 K-steps of 4 (N columns >=4 are zero) ----
        v2f B[5];
#pragma unroll
        for (int s = 0; s < 5; ++s) {
            const int k0 = 4 * s + (hi ? 2 : 0);
            float b0 = 0.0f, b1 = 0.0f;
            if (col < 4) {
                b0 = qm4_elem(x, k0,     col);
                b1 = qm4_elem(x, k0 + 1, col);
            }
            B[s].x = b0;
            B[s].y = b1;
        }

        // ---- A fragments: weights rows, two 16-node tiles per wave ----
        const int row0 = nodeBase + col;
        const float2* __restrict__ w0 =
            (const float2*)(weights + (size_t)row0 * KDIM);
        const float2* __restrict__ w1 =
            (const float2*)(weights + (size_t)(row0 + 16) * KDIM);

#pragma unroll
        for (int s = 0; s < 5; ++s) {
            const int f2off = 2 * s + (hi ? 1 : 0);   // K pair for this lane half
            const float2 a0 = w0[f2off];
            const float2 a1 = w1[f2off];
            v2f A0; A0.x = a0.x; A0.y = a0.y;
            v2f A1; A1.x = a1.x; A1.y = a1.y;
            c0 = __builtin_amdgcn_wmma_f32_16x16x4_f32(
                     false, A0, false, B[s], (short)0, c0, false, false);
            c1 = __builtin_amdgcn_wmma_f32_16x16x4_f32(
                     false, A1, false, B[s], (short)0, c1, false, false);
        }

        // ---- scatter the 4 useful columns (quaternion components) to LDS ----
        if (col < 4) {
#pragma unroll
            for (int v = 0; v < 8; ++v) {
                const int m = hi ? (v + 8) : v;       // row within 16-node tile
                qlds[wave][m][col]      = c0[v];
                qlds[wave][m + 16][col] = c1[v];
            }
        }
    }

    __syncthreads();

    if (active) {
        const int node = nodeBase + lane;

        float X = qlds[wave][lane][0];
        float Y = qlds[wave][lane][1];
        float Z = qlds[wave][lane][2];
        float W = qlds[wave][lane][3];
        const float inv = rsqrtf(X * X + Y * Y + Z * Z + W * W);
        X *= inv; Y *= inv; Z *= inv; W *= inv;

        const float4 vr = ((const float4*)VR)[node];
        const float v0 = vr.x, v1 = vr.y, v2 = vr.z;

        // t = H^T v   (H rows: (W,Z,-Y,-X), (-Z,W,X,-Y), (Y,-X,W,-Z))
        const float t0 =  W * v0 - Z * v1 + Y * v2;
        const float t1 =  Z * v0 + W * v1 - X * v2;
        const float t2 = -Y * v0 + X * v1 + W * v2;
        const float t3 = -X * v0 - Y * v1 - Z * v2;

        // y3 = G t    (G rows: (W,-Z,Y,-X), (Z,W,-X,-Y), (-Y,X,W,-Z))
        const float y0 =  W * t0 - Z * t1 + Y * t2 - X * t3;
        const float y1 =  Z * t0 + W * t1 - X * t2 - Y * t3;
        const float y2 = -Y * t0 + X * t1 + W * t2 - Z * t3;

        float4 yo;
        yo.x = y0; yo.y = y1; yo.z = y2; yo.w = vr.w;
        ((float4*)out)[node] = yo;
    }
}

extern "C" void kernel_launch(void* const* d_in, const int* in_sizes, int n_in,
                              void* d_out, int out_size, void* d_ws, size_t ws_size,
                              hipStream_t stream) {
    (void)in_sizes; (void)n_in; (void)out_size; (void)d_ws; (void)ws_size;
    const float* x       = (const float*)d_in[0];
    const float* weights = (const float*)d_in[1];
    const float* VR      = (const float*)d_in[2];
    float* out           = (float*)d_out;

    const int numBlocks = (N_NODES + 255) / 256;  // 7813; tail waves masked in-kernel
    dqs_wmma_kernel<<<numBlocks, 256, 0, stream>>>(x, weights, VR, out);
}